// AgMixPooler_1206_10548439679367
// MI455X (gfx1250) — compile-verified
//
#include <hip/hip_runtime.h>
#include <hip/hip_bf16.h>
#include <cstdint>

// Problem constants (match reference)
#define BB   32
#define TT   4096
#define EE   512
#define RR   64
#define WINN 7
#define KK   512

typedef float v2f __attribute__((ext_vector_type(2)));
typedef float v8f __attribute__((ext_vector_type(8)));

// TDM descriptor SGPR-group vector types (probe-confirmed builtin operand shapes)
typedef unsigned int tdm_v4u __attribute__((ext_vector_type(4)));
typedef int          tdm_v8i __attribute__((ext_vector_type(8)));
typedef int          tdm_v4i __attribute__((ext_vector_type(4)));

// ---------------------------------------------------------------------------
// TDM: DMA one contiguous 64x64 f32 W1 K-chunk (16KB) from global into LDS.
// D# built per CDNA5 ISA 8.3/8.4:
//   group0: count=1 | lds_addr | global_addr[56:0] | type=2
//   group1: data_size=4B, tensor_dim0/1=64, tile_dim0/1=64, dim0_stride=64
// Issued by one wave; completion tracked on TENSORcnt.
// ---------------------------------------------------------------------------
__device__ __forceinline__ void tdm_stage_w1(const float* W1, int c, unsigned ldsOff)
{
    unsigned long long ga = (unsigned long long)(uintptr_t)(W1 + (size_t)c * 64 * 64);
    tdm_v4u g0;
    g0.x = 1u;                                            // count=1, user mode
    g0.y = ldsOff;                                        // lds_addr (bytes)
    g0.z = (unsigned)ga;                                  // global_addr[31:0]
    g0.w = (unsigned)((ga >> 32) & 0x01FFFFFFull)         // global_addr[56:32]
           | (2u << 30);                                  // type = 2 ("image")
    tdm_v8i g1;
    g1[0] = 0x00020000;                                   // wg_mask=0, data_size=2 (4B)
    g1[1] = 64 << 16;                                     // tensor_dim0 = 64
    g1[2] = 64 << 16;                                     // tensor_dim1 = 64
    g1[3] = 64 << 16;                                     // tile_dim0   = 64
    g1[4] = 64;                                           // tile_dim1=64, tile_dim2=0
    g1[5] = 64;                                           // tensor_dim0_stride = 64
    g1[6] = 0;
    g1[7] = 0;
    tdm_v4i gz = (tdm_v4i)0;                              // groups 2/3 unused (2D)
#if defined(__clang_major__) && (__clang_major__ >= 23)
    __builtin_amdgcn_tensor_load_to_lds(g0, g1, gz, gz, (tdm_v8i)0, 0);
#else
    __builtin_amdgcn_tensor_load_to_lds(g0, g1, gz, gz, 0);
#endif
}

// ---------------------------------------------------------------------------
// Kernel 1: fused  hn = LayerNorm(GELU(embs @ W1 + b1)) * ln_g + ln_b
// One wave computes a 64(M) x 64(N) f32 tile with V_WMMA_F32_16X16X4_F32.
// W1 staged to LDS in 64-K chunks, double-buffered via TDM (wave 0 issues).
// ---------------------------------------------------------------------------
__global__ void __launch_bounds__(256)
gemm_gelu_ln_kernel(const float* __restrict__ embs, const float* __restrict__ W1,
                    const float* __restrict__ b1v_g, const float* __restrict__ lng,
                    const float* __restrict__ lnb, float* __restrict__ hn)
{
    __shared__ float w1s[2][64 * 64];   // 2 x 16KB double buffer of W1 K-chunks

    const int tid  = threadIdx.x;
    const int lane = tid & 31;
    const int wave = tid >> 5;
    const int half = lane >> 4;         // 0: lanes 0-15, 1: lanes 16-31
    const int l15  = lane & 15;
    const int rowW = blockIdx.x * 512 + wave * 64;   // 8 waves x 64 rows

    v8f acc[4][4];
#pragma unroll
    for (int mt = 0; mt < 4; ++mt)
#pragma unroll
        for (int n = 0; n < 4; ++n)
            acc[mt][n] = (v8f)0.0f;

    if (wave == 0) {
        tdm_stage_w1(W1, 0, (unsigned)(uintptr_t)(&w1s[0][0]));
        __builtin_amdgcn_s_wait_tensorcnt(0);
    }
    __syncthreads();

    for (int chunk = 0; chunk < 8; ++chunk) {
        const int buf = chunk & 1;
        if ((wave == 0) && (chunk < 7))                    // overlap DMA with math
            tdm_stage_w1(W1, chunk + 1, (unsigned)(uintptr_t)(&w1s[buf ^ 1][0]));

        for (int kk = 0; kk < 64; kk += 4) {
            const int kg = chunk * 64 + kk;
            v2f afrag[4];
#pragma unroll
            for (int mt = 0; mt < 4; ++mt) {
                // lanes l and l+16 fetch the same 16B (one line); split K halves
                const float4 av = *(const float4*)(
                    embs + (size_t)(rowW + mt * 16 + l15) * EE + kg);
                afrag[mt].x = half ? av.z : av.x;   // K = kk   | kk+2
                afrag[mt].y = half ? av.w : av.y;   // K = kk+1 | kk+3
            }
#pragma unroll
            for (int n = 0; n < 4; ++n) {
                const int col = n * 16 + l15;
                const int kr  = kk + 2 * half;
                v2f bfrag;
                bfrag.x = w1s[buf][kr * 64 + col];
                bfrag.y = w1s[buf][(kr + 1) * 64 + col];
#pragma unroll
                for (int mt = 0; mt < 4; ++mt)
                    acc[mt][n] = __builtin_amdgcn_wmma_f32_16x16x4_f32(
                        false, afrag[mt], false, bfrag, (short)0, acc[mt][n],
                        false, false);
            }
        }
        if (wave == 0) __builtin_amdgcn_s_wait_tensorcnt(0);
        __syncthreads();
    }

    // ---- epilogue: GELU + LayerNorm(R=64), rows live in 16-lane halves ----
    float b1r[4], gr[4], br[4];
#pragma unroll
    for (int n = 0; n < 4; ++n) {
        const int c = n * 16 + l15;
        b1r[n] = b1v_g[c]; gr[n] = lng[c]; br[n] = lnb[c];
    }

#pragma unroll
    for (int mt = 0; mt < 4; ++mt) {
#pragma unroll
        for (int v = 0; v < 8; ++v) {
            float g[4], s = 0.f, s2 = 0.f;
#pragma unroll
            for (int n = 0; n < 4; ++n) {
                const float x = acc[mt][n][v] + b1r[n];
                const float t = 0.5f * x * (1.0f + erff(x * 0.70710678118654752f));
                g[n] = t; s += t; s2 += t * t;
            }
#pragma unroll
            for (int m = 1; m < 16; m <<= 1) {       // reduce within 16-lane half
                s  += __shfl_xor(s,  m, 32);
                s2 += __shfl_xor(s2, m, 32);
            }
            const float mu  = s * (1.0f / 64.0f);
            const float var = s2 * (1.0f / 64.0f) - mu * mu;
            const float inv = rsqrtf(var + 1e-5f);
            const int row = rowW + mt * 16 + v + (half << 3);
#pragma unroll
            for (int n = 0; n < 4; ++n)
                hn[(size_t)row * RR + n * 16 + l15] = (g[n] - mu) * inv * gr[n] + br[n];
        }
    }
}

// ---------------------------------------------------------------------------
// Kernel 2: conv(7,R) over hn + ssf + sigmoid-gated tanh + mask -> a[B,T]
// hn window staged into padded LDS tile via GLOBAL_LOAD_ASYNC_TO_LDS_B128.
// ---------------------------------------------------------------------------
#define CT 128
__global__ void __launch_bounds__(CT)
conv_fuse_kernel(const float* __restrict__ hn, const float* __restrict__ ssfx,
                 const unsigned char* __restrict__ mask,
                 const float* __restrict__ convw, const float* __restrict__ convb,
                 const float* __restrict__ ssfw, const float* __restrict__ ssfb,
                 const float* __restrict__ gate, float* __restrict__ aout)
{
    __shared__ float tile[(CT + 6) * 68];   // stride 68 floats (16B aligned rows)
    __shared__ float cw[WINN * RR];

    const int b  = blockIdx.x / (TT / CT);
    const int t0 = (blockIdx.x % (TT / CT)) * CT;
    const int tid = threadIdx.x;
    const unsigned ldsBase = (unsigned)(uintptr_t)(&tile[0]);

    for (int i = tid; i < WINN * RR; i += CT) cw[i] = convw[i];

    for (int e = tid; e < (CT + 6) * 16; e += CT) {
        const int r  = e >> 4;
        const int c4 = (e & 15) * 4;
        const int t  = t0 - 3 + r;
        const unsigned dst = ldsBase + (unsigned)(r * 68 + c4) * 4u;
        if (t >= 0 && t < TT) {
            unsigned long long g = (unsigned long long)(uintptr_t)(
                hn + ((size_t)b * TT + t) * RR + c4);
            asm volatile("global_load_async_to_lds_b128 %0, %1, off"
                         :: "v"(dst), "v"(g) : "memory");
        } else {
            *(float4*)(tile + r * 68 + c4) = make_float4(0.f, 0.f, 0.f, 0.f);
        }
    }
    asm volatile("s_wait_asynccnt 0x0" ::: "memory");
    __syncthreads();

    const int t = t0 + tid;
    float wc = 0.f;
#pragma unroll
    for (int j = 0; j < WINN; ++j) {
        const float* rowp = tile + (tid + j) * 68;
        const float* wp   = cw + j * RR;
        for (int r = 0; r < RR; ++r) wc = fmaf(rowp[r], wp[r], wc);
    }
    wc += convb[0];

    const float* sx = ssfx + ((size_t)b * TT + t) * 7;
    float wssf = ssfb[0];
#pragma unroll
    for (int i = 0; i < 7; ++i) wssf = fmaf(sx[i], ssfw[i], wssf);

    const float alpha = 1.0f / (1.0f + expf(-gate[0]));
    float av = tanhf(alpha * wc + (1.0f - alpha) * wssf);
    if (!mask[(size_t)b * TT + t]) av = -INFINITY;
    aout[(size_t)b * TT + t] = av;
}

// ---------------------------------------------------------------------------
// Kernel 3: softmax over T per batch -> attn section of d_out
// ---------------------------------------------------------------------------
__global__ void __launch_bounds__(256)
softmax_kernel(const float* __restrict__ a, float* __restrict__ attn)
{
    __shared__ float red[256];
    const int b = blockIdx.x, tid = threadIdx.x;
    const float* ab = a + (size_t)b * TT;

    float mx = -INFINITY;
    for (int t = tid; t < TT; t += 256) mx = fmaxf(mx, ab[t]);
    red[tid] = mx; __syncthreads();
    for (int s = 128; s > 0; s >>= 1) {
        if (tid < s) red[tid] = fmaxf(red[tid], red[tid + s]);
        __syncthreads();
    }
    mx = red[0]; __syncthreads();

    float sum = 0.f;
    for (int t = tid; t < TT; t += 256) sum += expf(ab[t] - mx);
    red[tid] = sum; __syncthreads();
    for (int s = 128; s > 0; s >>= 1) {
        if (tid < s) red[tid] += red[tid + s];
        __syncthreads();
    }
    const float inv = 1.0f / red[0];
    for (int t = tid; t < TT; t += 256)
        attn[(size_t)b * TT + t] = expf(ab[t] - mx) * inv;
}

// ---------------------------------------------------------------------------
// Kernel 4: selection rank with lax.top_k tie rule (value desc, index asc)
// ---------------------------------------------------------------------------
__global__ void __launch_bounds__(256)
rank_kernel(const float* __restrict__ a, int* __restrict__ sel)
{
    __shared__ float as[TT];
    const int b   = blockIdx.x >> 4;
    const int seg = blockIdx.x & 15;
    for (int t = threadIdx.x; t < TT; t += 256) as[t] = a[(size_t)b * TT + t];
    __syncthreads();

    const int t = seg * 256 + threadIdx.x;
    const float my = as[t];
    int r = 0;
#pragma unroll 4
    for (int j = 0; j < TT; ++j) {
        const float v = as[j];
        r += (v > my) || ((v == my) && (j < t));
    }
    sel[(size_t)b * TT + t] = (r < KK) ? 1 : 0;
}

// ---------------------------------------------------------------------------
// Kernel 5: per-batch prefix-scan of flags -> compacted ascending indices
// ---------------------------------------------------------------------------
__global__ void __launch_bounds__(256)
compact_kernel(const int* __restrict__ sel, int* __restrict__ idx)
{
    __shared__ int sc[256];
    const int b = blockIdx.x, tid = threadIdx.x;
    const int* sb = sel + (size_t)b * TT;
    const int base = tid * 16;
    int c = 0;
#pragma unroll
    for (int i = 0; i < 16; ++i) c += sb[base + i];
    sc[tid] = c; __syncthreads();
    for (int off = 1; off < 256; off <<= 1) {
        int v = (tid >= off) ? sc[tid - off] : 0;
        __syncthreads();
        sc[tid] += v;
        __syncthreads();
    }
    int pos = sc[tid] - c;                  // exclusive prefix
#pragma unroll
    for (int i = 0; i < 16; ++i) {
        if (sb[base + i] && pos < KK) { idx[(size_t)b * KK + pos] = base + i; ++pos; }
    }
}

// ---------------------------------------------------------------------------
// Kernel 6: gather selected rows -> pooled section of d_out
// ---------------------------------------------------------------------------
__global__ void __launch_bounds__(128)
gather_kernel(const float* __restrict__ embs, const int* __restrict__ idx,
              float* __restrict__ pooled)
{
    const int bk = blockIdx.x;
    const int b  = bk / KK;
    const int t  = idx[bk];
    const float4* src = (const float4*)(embs + ((size_t)b * TT + t) * EE);
    float4*       dst = (float4*)(pooled + (size_t)bk * EE);
    dst[threadIdx.x] = src[threadIdx.x];    // 128 threads x 16B = 512 floats
}

// ---------------------------------------------------------------------------
extern "C" void kernel_launch(void* const* d_in, const int* in_sizes, int n_in,
                              void* d_out, int out_size, void* d_ws, size_t ws_size,
                              hipStream_t stream)
{
    const float*         embs  = (const float*)d_in[0];
    const float*         ssfx  = (const float*)d_in[1];
    const unsigned char* maskp = (const unsigned char*)d_in[2];
    const float*         W1    = (const float*)d_in[3];
    const float*         b1    = (const float*)d_in[4];
    const float*         lng   = (const float*)d_in[5];
    const float*         lnb   = (const float*)d_in[6];
    const float*         convw = (const float*)d_in[7];
    const float*         convb = (const float*)d_in[8];
    const float*         ssfw  = (const float*)d_in[9];
    const float*         ssfb  = (const float*)d_in[10];
    const float*         gate  = (const float*)d_in[11];

    float* pooled = (float*)d_out;                        // [B,K,E]
    float* attn   = pooled + (size_t)BB * KK * EE;        // [B,T,1]

    // workspace layout (~34.7 MB)
    float* hn   = (float*)d_ws;                           // [B,T,R]
    float* aarr = hn + (size_t)BB * TT * RR;              // [B,T]
    int*   sel  = (int*)(aarr + (size_t)BB * TT);         // [B,T]
    int*   idx  = sel + (size_t)BB * TT;                  // [B,K]

    gemm_gelu_ln_kernel<<<(BB * TT) / 512, 256, 0, stream>>>(embs, W1, b1, lng, lnb, hn);
    conv_fuse_kernel<<<BB * (TT / CT), CT, 0, stream>>>(hn, ssfx, maskp, convw, convb,
                                                        ssfw, ssfb, gate, aarr);
    softmax_kernel<<<BB, 256, 0, stream>>>(aarr, attn);
    rank_kernel<<<BB * 16, 256, 0, stream>>>(aarr, sel);
    compact_kernel<<<BB, 256, 0, stream>>>(sel, idx);
    gather_kernel<<<BB * KK, 128, 0, stream>>>(embs, idx, pooled);
}